// Filter_34522947125484
// MI455X (gfx1250) — compile-verified
//
#include <hip/hip_runtime.h>

#define BATCH 256
#define SEQ   4096
#define NS    16
#define MS    8
#define US    4
#define STEPF 448   // floats per step in workspace: AeffT(256) + BeffT(64) + K(128)

typedef __attribute__((ext_vector_type(2))) float v2f;
typedef __attribute__((ext_vector_type(8))) float v8f;

// ---------------------------------------------------------------------------
// Phase 1: batch-independent covariance recursion (sequential over time).
// One block of 256 threads cooperatively does the small dense algebra in LDS
// and streams per-step effective matrices to the workspace:
//   ws[t-1] = { AeffT = (I-KC)A : 16x16, BeffT = (I-KC)B : 16x4, K : 16x8 }
// ---------------------------------------------------------------------------
__global__ __launch_bounds__(256)
void kalman_cov_kernel(const float* __restrict__ Ag, const float* __restrict__ Bg,
                       const float* __restrict__ Cg, const float* __restrict__ Qg,
                       const float* __restrict__ Rg, float* __restrict__ ws)
{
    __shared__ float sA[NS*NS], sB[NS*US], sC[MS*NS], sQ[NS*NS], sR[MS*MS];
    __shared__ float sP[NS*NS], sT1[NS*NS], sPp[NS*NS];
    __shared__ float sCP[MS*NS], sSI[MS][2*MS], sPCt[NS*MS], sK[NS*MS], sW[NS*NS];

    const int tid = threadIdx.x;
    sA[tid] = Ag[tid];
    sQ[tid] = Qg[tid];
    if (tid < NS*US) sB[tid] = Bg[tid];
    if (tid < MS*NS) sC[tid] = Cg[tid];
    if (tid < MS*MS) sR[tid] = Rg[tid];
    {
        int i = tid >> 4, j = tid & 15;
        sP[tid] = (i == j) ? 1.0f : 0.0f;   // P0 = eye(n)
    }
    __syncthreads();

    const int i16 = tid >> 4, j16 = tid & 15;   // 16x16 / 8x16 mapping
    const int i8  = tid >> 3, j8  = tid & 7;    // 16x8 mapping (tid<128)

    #pragma unroll 1
    for (int t = 1; t < SEQ; ++t) {
        // T1 = A @ P
        float acc = 0.f;
        #pragma unroll
        for (int k = 0; k < NS; ++k) acc += sA[i16*NS + k] * sP[k*NS + j16];
        sT1[tid] = acc;
        __syncthreads();
        // Ppri = T1 @ A^T + Q
        acc = sQ[tid];
        #pragma unroll
        for (int k = 0; k < NS; ++k) acc += sT1[i16*NS + k] * sA[j16*NS + k];
        sPp[tid] = acc;
        __syncthreads();
        // CP = C @ Ppri (8x16)   and   PCt = Ppri @ C^T (16x8)
        if (tid < MS*NS) {
            float a = 0.f;
            #pragma unroll
            for (int k = 0; k < NS; ++k) a += sC[i16*NS + k] * sPp[k*NS + j16];
            sCP[i16*NS + j16] = a;
            float b = 0.f;
            #pragma unroll
            for (int k = 0; k < NS; ++k) b += sPp[i8*NS + k] * sC[j8*NS + k];
            sPCt[i8*MS + j8] = b;
        }
        __syncthreads();
        // Augmented [S | I],  S = CP @ C^T + R  (8x8, SPD)
        if (tid < MS*NS) {
            float v;
            if (j16 < MS) {
                v = sR[i16*MS + j16];
                #pragma unroll
                for (int k = 0; k < NS; ++k) v += sCP[i16*NS + k] * sC[j16*NS + k];
            } else {
                v = ((j16 - MS) == i16) ? 1.0f : 0.0f;
            }
            sSI[i16][j16] = v;
        }
        __syncthreads();
        // Gauss-Jordan inverse of S (no pivot search: S is SPD, R = 0.1*I)
        #pragma unroll 1
        for (int p = 0; p < MS; ++p) {
            float invp = 1.0f / sSI[p][p];
            float pv = 0.f, f = 0.f;
            if (tid < MS*NS) { pv = sSI[p][j16] * invp; f = sSI[i16][p]; }
            __syncthreads();
            if (tid < MS*NS) sSI[i16][j16] = (i16 == p) ? pv : sSI[i16][j16] - f * pv;
            __syncthreads();
        }
        // K = PCt @ Sinv  (16x8)
        if (tid < NS*MS) {
            float v = 0.f;
            #pragma unroll
            for (int k = 0; k < MS; ++k) v += sPCt[i8*MS + k] * sSI[k][MS + j8];
            sK[i8*MS + j8] = v;
        }
        __syncthreads();
        // W = I - K @ C  (16x16)
        {
            float v = (i16 == j16) ? 1.0f : 0.0f;
            #pragma unroll
            for (int k = 0; k < MS; ++k) v -= sK[i16*MS + k] * sC[k*NS + j16];
            sW[tid] = v;
        }
        __syncthreads();
        float* wsp = ws + (size_t)(t - 1) * STEPF;
        // AeffT = W @ A  -> ws[0..255]
        {
            float v = 0.f;
            #pragma unroll
            for (int k = 0; k < NS; ++k) v += sW[i16*NS + k] * sA[k*NS + j16];
            wsp[tid] = v;
        }
        // P_post = W @ Ppri  (only read next iteration; barrier below)
        {
            float v = 0.f;
            #pragma unroll
            for (int k = 0; k < NS; ++k) v += sW[i16*NS + k] * sPp[k*NS + j16];
            sP[tid] = v;
        }
        // BeffT = W @ B (16x4) -> ws[256..319],  K -> ws[320..447]
        if (tid < NS*US) {
            int ib = tid >> 2, jb = tid & 3;
            float v = 0.f;
            #pragma unroll
            for (int k = 0; k < NS; ++k) v += sW[ib*NS + k] * sB[k*US + jb];
            wsp[256 + tid] = v;
        }
        if (tid < NS*MS) wsp[320 + tid] = sK[tid];
        __syncthreads();
    }
}

// ---------------------------------------------------------------------------
// Phase 2: per-batch affine state recursion with V_WMMA_F32_16X16X4_F32.
// State kept as X^T (state x batch) in WMMA C/D f32 layout; per step:
//   X^T <- AeffT @ X^T + (BeffT @ U^T + K @ Z^T)
// The parenthesized term (dzu) is state-independent -> computed one step
// ahead and fed as the C operand of the first recurrent WMMA, so the serial
// critical path per step is only: ds_bpermute relayout -> 4 chained WMMAs.
// Operand loads are software-pipelined one step ahead.
// 16 blocks x 1 wave (wave32), one 16-batch tile per wave.
// ---------------------------------------------------------------------------
__device__ __forceinline__ float bperm(int srcLane, float v) {
    return __builtin_bit_cast(float,
        __builtin_amdgcn_ds_bpermute(srcLane << 2, __builtin_bit_cast(int, v)));
}

__device__ __forceinline__ v8f wmma4(v2f a, v2f b, v8f c) {
    return __builtin_amdgcn_wmma_f32_16x16x4_f32(false, a, false, b,
                                                 (short)0, c, false, false);
}

struct StepOps {
    v2f aA0, aA1, aA2, aA3;   // AeffT chunks (A operand, 16x4 each)
    v2f aB, aK0, aK1;         // BeffT, K chunks (A operand)
    v2f bZ0, bZ1, bU;         // Z^T, U^T chunks (B operand)
};

__device__ __forceinline__ StepOps load_ops(const float* __restrict__ ws,
                                            const float* __restrict__ zbase,
                                            const float* __restrict__ ubase,
                                            int t, int col, int hi) {
    StepOps o;
    const float* w = ws + (size_t)(t - 1) * STEPF;
    // A-operand (16xK f32) layout: lane L -> M=L&15, {v0,v1}={K=2*hi, 2*hi+1} per 4-chunk
    o.aA0 = *(const v2f*)(w + col * 16 + 0  + 2 * hi);
    o.aA1 = *(const v2f*)(w + col * 16 + 4  + 2 * hi);
    o.aA2 = *(const v2f*)(w + col * 16 + 8  + 2 * hi);
    o.aA3 = *(const v2f*)(w + col * 16 + 12 + 2 * hi);
    o.aB  = *(const v2f*)(w + 256 + col * 4 + 2 * hi);
    o.aK0 = *(const v2f*)(w + 320 + col * 8 + 0 + 2 * hi);
    o.aK1 = *(const v2f*)(w + 320 + col * 8 + 4 + 2 * hi);
    // B-operand (Kx16) layout: lane L -> N=L&15, {v0,v1}={K=2*hi, 2*hi+1} per 4-chunk
    const float* zr = zbase + (size_t)t * MS;
    o.bZ0 = *(const v2f*)(zr + 0 + 2 * hi);
    o.bZ1 = *(const v2f*)(zr + 4 + 2 * hi);
    o.bU  = *(const v2f*)(ubase + (size_t)t * US + 2 * hi);
    return o;
}

// State-independent partial sum: BeffT @ U^T + K @ Z^T  (3 WMMAs, off path)
__device__ __forceinline__ v8f zu_chain(const StepOps& o) {
    v8f d = {};
    d = wmma4(o.aB,  o.bU,  d);
    d = wmma4(o.aK0, o.bZ0, d);
    d = wmma4(o.aK1, o.bZ1, d);
    return d;
}

__global__ __launch_bounds__(32)
void kalman_state_kernel(const float* __restrict__ obs, const float* __restrict__ inp,
                         const float* __restrict__ x0, const float* __restrict__ ws,
                         float* __restrict__ out)
{
    const int lane = threadIdx.x;
    const int hi   = lane >> 4;       // half-wave select
    const int col  = lane & 15;       // batch column within tile / A-operand row
    const int b    = blockIdx.x * 16 + col;

    // init accumulator: acc[r] = X^T[r + 8*hi][col] = x0[r + 8*hi]  (C/D layout)
    v8f x;
    #pragma unroll
    for (int r = 0; r < 8; ++r) x[r] = x0[8 * hi + r];

    // states[:, 0, :] = x0 (per lane: 8 contiguous floats)
    {
        float* o = out + ((size_t)b * SEQ) * NS + 8 * hi;
        *(float4*)(o)     = make_float4(x[0], x[1], x[2], x[3]);
        *(float4*)(o + 4) = make_float4(x[4], x[5], x[6], x[7]);
    }

    const float* zbase = obs + (size_t)b * SEQ * MS;
    const float* ubase = inp + (size_t)b * SEQ * US;
    float*       obase = out + (size_t)b * SEQ * NS;

    // Pipeline prologue: operands + state-independent partial sum for t=1
    StepOps cur = load_ops(ws, zbase, ubase, 1, col, hi);
    v8f dzu = zu_chain(cur);

    #pragma unroll 1
    for (int t = 1; t < SEQ; ++t) {
        // Issue next step's loads early (clamped; last iteration is redundant)
        const int tn = (t + 1 < SEQ) ? (t + 1) : t;
        StepOps nxt = load_ops(ws, zbase, ubase, tn, col, hi);

        // Convert accumulator (C/D layout: x[r]@lane = X^T[r+8*hi][lane&15])
        // into B-operand chunks bX[c]: need X^T[4c+2*hi+j][col] at this lane.
        v2f bX[4];
        #pragma unroll
        for (int c = 0; c < 4; ++c) {
            int srcLane = col + ((c >= 2) ? 16 : 0);
            int lo = 4 * (c & 1);
            float t0 = bperm(srcLane, x[lo + 0]);
            float t1 = bperm(srcLane, x[lo + 2]);
            float t2 = bperm(srcLane, x[lo + 1]);
            float t3 = bperm(srcLane, x[lo + 3]);
            bX[c].x = hi ? t1 : t0;
            bX[c].y = hi ? t3 : t2;
        }

        // Recurrent chain: X^T_new = AeffT @ X^T + dzu   (4 chained WMMAs)
        v8f d = wmma4(cur.aA0, bX[0], dzu);
        d = wmma4(cur.aA1, bX[1], d);
        d = wmma4(cur.aA2, bX[2], d);
        d = wmma4(cur.aA3, bX[3], d);
        x = d;

        // store states[b, t, :] : per lane 8 contiguous floats (s = 8*hi + r)
        float* o = obase + (size_t)t * NS + 8 * hi;
        *(float4*)(o)     = make_float4(x[0], x[1], x[2], x[3]);
        *(float4*)(o + 4) = make_float4(x[4], x[5], x[6], x[7]);

        // Off-critical-path: next step's state-independent partial sum
        dzu = zu_chain(nxt);
        cur = nxt;
    }
}

extern "C" void kernel_launch(void* const* d_in, const int* in_sizes, int n_in,
                              void* d_out, int out_size, void* d_ws, size_t ws_size,
                              hipStream_t stream) {
    const float* obs = (const float*)d_in[0];
    const float* inp = (const float*)d_in[1];
    const float* A   = (const float*)d_in[2];
    const float* B   = (const float*)d_in[3];
    const float* C   = (const float*)d_in[4];
    const float* Q   = (const float*)d_in[5];
    const float* R   = (const float*)d_in[6];
    const float* x0  = (const float*)d_in[7];
    float* out = (float*)d_out;
    float* ws  = (float*)d_ws;   // needs (SEQ-1)*448*4 ~= 7.34 MB

    hipLaunchKernelGGL(kalman_cov_kernel, dim3(1), dim3(256), 0, stream,
                       A, B, C, Q, R, ws);
    hipLaunchKernelGGL(kalman_state_kernel, dim3(BATCH / 16), dim3(32), 0, stream,
                       obs, inp, x0, ws, out);
}